// yoloLoss_35570919146025
// MI455X (gfx1250) — compile-verified
//
#include <hip/hip_runtime.h>
#include <hip/hip_bf16.h>
#include <stdint.h>

// ---------------------------------------------------------------------------
// YOLO loss for MI455X (gfx1250). Bandwidth-bound streaming reduction:
//  - input (64,90,56,56) f32 read exactly once, lane-coalesced (lane = i)
//  - per-batch GT table (20 boxes, 800 B) staged to LDS via CDNA5 async
//    global->LDS DMA (global_load_async_to_lds_b32 / s_wait_asynccnt)
//  - wave32 shuffle reduction per block, f64 global atomic accumulation
// ---------------------------------------------------------------------------

#define BS    64
#define NBOX  20
#define NC    80
#define GRID  56
#define CH    (GRID * GRID)          // 3136 elements per channel plane
#define CELLS_PER_BLOCK 128
#define BLOCKS_PER_BATCH 25          // ceil(3136 / 128)
#define IGNORE_T 0.8f
#define EPS_F 1e-7f

struct GT {                          // 40 bytes, 10 dwords
    float gx, gy, gw, gh, bsx, bsy;
    int   gi, gj, bn, cls;
};
#define GT_DWORDS_PER_BATCH (NBOX * 10)   // 200 dwords = 800 B

__device__ __forceinline__ float sigm(float x) { return 1.0f / (1.0f + expf(-x)); }
__device__ __forceinline__ float clampp(float p) {
    return fminf(fmaxf(p, EPS_F), 1.0f - EPS_F);
}

// CIoU between pred box (px,py,pw,ph) and target box (gx,gy,gw,gh), grid units.
// Mirrors the reference _box_ciou term-by-term.
__device__ __forceinline__ float ciou_fn(float px, float py, float pw, float ph,
                                         float gx, float gy, float gw, float gh) {
    float p1x = px - 0.5f * pw, p2x = px + 0.5f * pw;
    float p1y = py - 0.5f * ph, p2y = py + 0.5f * ph;
    float g1x = gx - 0.5f * gw, g2x = gx + 0.5f * gw;
    float g1y = gy - 0.5f * gh, g2y = gy + 0.5f * gh;
    float iwx = fmaxf(fminf(p2x, g2x) - fmaxf(p1x, g1x), 0.0f);
    float iwy = fmaxf(fminf(p2y, g2y) - fmaxf(p1y, g1y), 0.0f);
    float inter = iwx * iwy;
    float uni   = pw * ph + gw * gh - inter;
    float iou   = inter / fmaxf(uni, 1e-6f);
    float cd    = (px - gx) * (px - gx) + (py - gy) * (py - gy);
    float ewx   = fmaxf(fmaxf(p2x, g2x) - fminf(p1x, g1x), 0.0f);
    float ewy   = fmaxf(fmaxf(p2y, g2y) - fminf(p1y, g1y), 0.0f);
    float ed    = ewx * ewx + ewy * ewy;
    float c     = iou - cd / fmaxf(ed, 1e-6f);
    float dv    = atanf(pw / fmaxf(ph, 1e-6f)) - atanf(gw / fmaxf(gh, 1e-6f));
    float v     = 0.4052847345693511f * dv * dv;      // (4/pi^2) * dv^2
    float alpha = v / fmaxf(1.0f - iou + v, 1e-6f);
    return c - alpha * v;
}

// ---------------------------------------------------------------------------
// Kernel 1: per-GT-box prep (1280 boxes) + zero the f64 accumulator.
// ---------------------------------------------------------------------------
__global__ void __launch_bounds__(128)
k_gt_init(const float* __restrict__ tgt, GT* __restrict__ gt,
          double* __restrict__ accum) {
    int idx = blockIdx.x * blockDim.x + threadIdx.x;
    if (idx == 0) *accum = 0.0;
    if (idx >= BS * NBOX) return;
    const float* t = tgt + (size_t)idx * 5;
    float x1 = t[0], y1 = t[1], x2 = t[2], y2 = t[3];
    int cls = (int)t[4];
    float wx = x2 - x1, wy = y2 - y1;
    float gx = 0.5f * (x1 + x2) * (float)GRID;
    float gy = 0.5f * (y1 + y2) * (float)GRID;
    float gw = wx * (float)GRID;
    float gh = wy * (float)GRID;
    int gi = min(max((int)floorf(gx), 0), GRID - 1);
    int gj = min(max((int)floorf(gy), 0), GRID - 1);
    // anchors / stride(=8): a0 = (3.5, 20.0), a1 = (17.125, 47.875)
    float area = gw * gh;
    float ia0 = fminf(gw, 3.5f)    * fminf(gh, 20.0f);
    float ia1 = fminf(gw, 17.125f) * fminf(gh, 47.875f);
    float r0 = ia0 / (area + 70.0f       - ia0);
    float r1 = ia1 / (area + 819.859375f - ia1);
    int bn = (r1 > r0) ? 1 : 0;   // argmax: first index wins ties
    GT g;
    g.gx = gx; g.gy = gy; g.gw = gw; g.gh = gh; g.bsx = wx; g.bsy = wy;
    g.gi = gi; g.gj = gj; g.bn = bn; g.cls = cls;
    gt[idx] = g;
}

// ---------------------------------------------------------------------------
// Kernel 2: fused main loss. One thread per (b, j, i) cell:
//   both anchors' conf/loc terms + 80-class BCE. Block = 128 cells of one
//   batch; 25 blocks per batch (last block partially idle).
// ---------------------------------------------------------------------------
__global__ void __launch_bounds__(CELLS_PER_BLOCK)
k_main(const float* __restrict__ in, const GT* __restrict__ gt,
       double* __restrict__ accum) {
    __shared__ GT    shGT[NBOX];
    __shared__ float shRed[CELLS_PER_BLOCK / 32];

    const int tid   = threadIdx.x;
    const int b     = blockIdx.x / BLOCKS_PER_BATCH;
    const int local = (blockIdx.x % BLOCKS_PER_BATCH) * CELLS_PER_BLOCK + tid;

    // --- Stage this batch's 800 B GT table into LDS via async global->LDS ---
    {
        const uint32_t* gsrc = (const uint32_t*)(gt + b * NBOX);
        uint32_t*       sdst = (uint32_t*)shGT;
        for (int k = tid; k < GT_DWORDS_PER_BATCH; k += CELLS_PER_BLOCK) {
            uint32_t lds_off = (uint32_t)(uintptr_t)(sdst + k); // low 32 bits of
            const uint32_t* gp = gsrc + k;                      // LDS aperture
            asm volatile("global_load_async_to_lds_b32 %0, %1, off"
                         :: "v"(lds_off), "v"(gp) : "memory");
        }
        asm volatile("s_wait_asynccnt 0x0" ::: "memory");
        __syncthreads();
    }

    float acc = 0.0f;
    if (local < CH) {
        const int j = local / GRID;
        const int i = local % GRID;
        const float* base = in + (size_t)b * 90 * CH + local;

        float pch[10];
#pragma unroll
        for (int k = 0; k < 10; ++k) pch[k] = base[k * CH];

        // Decode both anchor boxes.
        const float sax[2] = {3.5f, 17.125f};
        const float say[2] = {20.0f, 47.875f};
        float px[2], py[2], pw[2], ph[2], pconf[2];
#pragma unroll
        for (int n = 0; n < 2; ++n) {
            px[n]    = sigm(pch[5 * n + 0]) + (float)i;
            py[n]    = sigm(pch[5 * n + 1]) + (float)j;
            pw[n]    = expf(pch[5 * n + 2]) * sax[n];
            ph[n]    = expf(pch[5 * n + 3]) * say[n];
            pconf[n] = sigm(pch[5 * n + 4]);
        }

        // Single pass over the GT table: best IoU + assignment per anchor,
        // plus the class-target bitmask for this cell.
        float best[2] = {0.0f, 0.0f};
        int   asg[2]  = {-1, -1};
        uint32_t cm0 = 0, cm1 = 0, cm2 = 0;
        for (int t = 0; t < NBOX; ++t) {
            GT g = shGT[t];
            float g1x = g.gx - 0.5f * g.gw, g2x = g.gx + 0.5f * g.gw;
            float g1y = g.gy - 0.5f * g.gh, g2y = g.gy + 0.5f * g.gh;
            float garea = g.gw * g.gh;
#pragma unroll
            for (int n = 0; n < 2; ++n) {
                float iwx = fmaxf(fminf(px[n] + 0.5f * pw[n], g2x) -
                                  fmaxf(px[n] - 0.5f * pw[n], g1x), 0.0f);
                float iwy = fmaxf(fminf(py[n] + 0.5f * ph[n], g2y) -
                                  fmaxf(py[n] - 0.5f * ph[n], g1y), 0.0f);
                float inter = iwx * iwy;
                float u     = pw[n] * ph[n] + garea - inter;
                best[n] = fmaxf(best[n], inter / u);
                if (g.bn == n && g.gj == j && g.gi == i) asg[n] = t; // last wins
            }
            if (g.gj == j && g.gi == i) {
                int c = g.cls;
                if      (c < 32) cm0 |= 1u << c;
                else if (c < 64) cm1 |= 1u << (c - 32);
                else             cm2 |= 1u << (c - 64);
            }
        }

        // Conf + loc losses.
        float conf_loss = 0.0f, loc_loss = 0.0f;
#pragma unroll
        for (int n = 0; n < 2; ++n) {
            float p = clampp(pconf[n]);
            if (asg[n] >= 0) {
                conf_loss += -logf(p);              // BCE(p, 1) at object cell
                GT g = shGT[asg[n]];
                float cio = ciou_fn(px[n], py[n], pw[n], ph[n],
                                    g.gx, g.gy, g.gw, g.gh);
                loc_loss += (2.0f - g.bsx * g.bsy) * (1.0f - cio);
            } else if (!(best[n] > IGNORE_T)) {
                conf_loss += -logf(1.0f - p);       // BCE(p, 0) on noobj
            }
        }

        // Class loss over 80 channels (sparse one-hot target).
        float cls_loss = 0.0f;
        const float* cbase = base + 10 * CH;
#pragma unroll 4
        for (int c = 0; c < NC; ++c) {
            float z = cbase[c * CH];
            float p = clampp(sigm(z));
            uint32_t bit = ((c < 32 ? (cm0 >> c)
                                    : c < 64 ? (cm1 >> (c - 32))
                                             : (cm2 >> (c - 64))) & 1u);
            cls_loss += bit ? -logf(p) : -logf(1.0f - p);
        }

        acc = (4.0f * conf_loss + 3.0f * cls_loss + loc_loss) * (1.0f / (float)BS);
    }

    // --- wave32 shuffle reduction, then cross-wave via LDS ---
#pragma unroll
    for (int off = 16; off > 0; off >>= 1)
        acc += __shfl_down(acc, off, 32);
    const int wid = tid >> 5, lid = tid & 31;
    if (lid == 0) shRed[wid] = acc;
    __syncthreads();
    if (tid == 0) {
        float s = 0.0f;
#pragma unroll
        for (int w = 0; w < CELLS_PER_BLOCK / 32; ++w) s += shRed[w];
        atomicAdd(accum, (double)s);   // f64 atomic: negligible ordering error
    }
}

// ---------------------------------------------------------------------------
// Kernel 3: scalar writeback (f64 accumulator -> f32 output).
// ---------------------------------------------------------------------------
__global__ void k_fin(const double* __restrict__ accum, float* __restrict__ out) {
    out[0] = (float)(*accum);
}

extern "C" void kernel_launch(void* const* d_in, const int* in_sizes, int n_in,
                              void* d_out, int out_size, void* d_ws, size_t ws_size,
                              hipStream_t stream) {
    const float* x   = (const float*)d_in[0];   // (64, 90, 56, 56) f32
    const float* tgt = (const float*)d_in[1];   // (64, 20, 5) f32
    float*  out   = (float*)d_out;              // scalar f32
    double* accum = (double*)d_ws;              // ws[0..15]: f64 accumulator
    GT*     gt    = (GT*)((char*)d_ws + 16);    // ws[16..]: 1280 * 40 B

    k_gt_init<<<(BS * NBOX + 127) / 128, 128, 0, stream>>>(tgt, gt, accum);
    k_main<<<BS * BLOCKS_PER_BATCH, CELLS_PER_BLOCK, 0, stream>>>(x, gt, accum);
    k_fin<<<1, 1, 0, stream>>>(accum, out);
}